// SetConv1dEncoder_858993459735
// MI455X (gfx1250) — compile-verified
//
#include <hip/hip_runtime.h>

// ---------------------------------------------------------------------------
// SetConv1dEncoder for MI455X (gfx1250, wave32).
//   out^T[b, c, m] = sum_n z_aug[b, n, c] * w[b, m, n]
//   w computed on the fly (Gaussian), density channel via lane-sum of w.
// Matrix core: V_WMMA_F32_16X16X4_F32 (fp32 in/out, matches reference dtype).
// ---------------------------------------------------------------------------

#define BATCH   8
#define N_CTX   2048
#define N_GRID  8192
#define CH      64          // z channels (z_aug has CH+1 = 65)
#define WAVES   8
#define BLOCK   (WAVES * 32)

typedef __attribute__((ext_vector_type(2))) float v2f;
typedef __attribute__((ext_vector_type(8))) float v8f;

__global__ __launch_bounds__(BLOCK, 2)
void setconv1d_encoder_kernel(const float* __restrict__ xz,
                              const float* __restrict__ z,
                              const float* __restrict__ x_grid,
                              const float* __restrict__ log_scale,
                              float* __restrict__ out)
{
    __shared__ float s_xz[N_CTX];   // 8 KB: all context coords for this batch

    const int b   = blockIdx.y;
    const int tid = threadIdx.x;

    // Cooperatively stage xz[b, :] into LDS (read N_CTX/BLOCK = 8 per thread).
    #pragma unroll
    for (int i = tid; i < N_CTX; i += BLOCK)
        s_xz[i] = xz[b * N_CTX + i];
    __syncthreads();

    const int wave = tid >> 5;
    const int lane = tid & 31;
    const int l15  = lane & 15;
    const int hi   = lane >> 4;                       // half-wave: 0 or 1
    const int m0   = (blockIdx.x * WAVES + wave) * 16; // grid-tile start

    // Per-lane grid coordinate (column n = l15 of the B tile).
    const float xg   = x_grid[(size_t)b * N_GRID + m0 + l15];
    // w = exp(-0.5 * d2 / exp(2*ls)) = exp(coef * d2)
    const float coef = -0.5f * __expf(-2.0f * log_scale[0]);

    const float* zb = z + (size_t)b * N_CTX * CH;

    v8f acc0 = {}, acc1 = {}, acc2 = {}, acc3 = {};
    float dens = 0.0f;   // partial row-sum of w (density channel)

    #pragma unroll 2
    for (int k = 0; k < N_CTX; k += 4) {
        // ---- B tile: w^T (4x16). Lane holds rows K = k+2*hi, k+2*hi+1 ----
        const int   kr  = k + 2 * hi;
        const float dx0 = xg - s_xz[kr];
        const float dx1 = xg - s_xz[kr + 1];
        const float w0  = __expf(coef * dx0 * dx0);
        const float w1  = __expf(coef * dx1 * dx1);
        dens += w0 + w1;
        v2f bm; bm.x = w0; bm.y = w1;

        // ---- A tiles: z^T (16x4). Lane: M = channel = l15 (+16*tile),
        //      K = kr (vgpr0), kr+1 (vgpr1). Lanes 0-15 -> 64B coalesced. ----
        const float* zr0 = zb + (size_t)kr * CH + l15;
        const float* zr1 = zr0 + CH;
        v2f a0, a1, a2, a3;
        a0.x = zr0[0];  a0.y = zr1[0];
        a1.x = zr0[16]; a1.y = zr1[16];
        a2.x = zr0[32]; a2.y = zr1[32];
        a3.x = zr0[48]; a3.y = zr1[48];

        // D = A x B + C  (16x16 f32 accumulators, one per channel tile)
        acc0 = __builtin_amdgcn_wmma_f32_16x16x4_f32(false, a0, false, bm,
                                                     (short)0, acc0, false, false);
        acc1 = __builtin_amdgcn_wmma_f32_16x16x4_f32(false, a1, false, bm,
                                                     (short)0, acc1, false, false);
        acc2 = __builtin_amdgcn_wmma_f32_16x16x4_f32(false, a2, false, bm,
                                                     (short)0, acc2, false, false);
        acc3 = __builtin_amdgcn_wmma_f32_16x16x4_f32(false, a3, false, bm,
                                                     (short)0, acc3, false, false);
    }

    // Full density: combine the two half-wave K partitions.
    dens += __shfl_xor(dens, 16, 32);
    const float inv = 1.0f / (dens + 1e-8f);

    // ---- Store, output layout (b, 65, N_GRID) ----
    float* ob = out + (size_t)b * (CH + 1) * N_GRID;

    // Channels 1..64: D layout -> vgpr v, lane: row c = v + 8*hi, col = l15.
    #pragma unroll
    for (int v = 0; v < 8; ++v) {
        const size_t row = (size_t)(1 + v + 8 * hi);
        const size_t col = (size_t)(m0 + l15);
        ob[(row +  0) * N_GRID + col] = acc0[v] * inv;
        ob[(row + 16) * N_GRID + col] = acc1[v] * inv;
        ob[(row + 32) * N_GRID + col] = acc2[v] * inv;
        ob[(row + 48) * N_GRID + col] = acc3[v] * inv;
    }

    // Channel 0: density (lanes 0-15 hold unique grid columns).
    if (lane < 16)
        ob[m0 + lane] = dens;
}

extern "C" void kernel_launch(void* const* d_in, const int* in_sizes, int n_in,
                              void* d_out, int out_size, void* d_ws, size_t ws_size,
                              hipStream_t stream)
{
    const float* xz        = (const float*)d_in[0];
    const float* z         = (const float*)d_in[1];
    const float* x_grid    = (const float*)d_in[2];
    const float* log_scale = (const float*)d_in[3];
    float*       out       = (float*)d_out;

    dim3 grid(N_GRID / (16 * WAVES), BATCH);   // 64 x 8 blocks
    setconv1d_encoder_kernel<<<grid, BLOCK, 0, stream>>>(xz, z, x_grid,
                                                         log_scale, out);
}